// ParallelMultiScaleAggregation_88192858456452
// MI455X (gfx1250) — compile-verified
//
#include <hip/hip_runtime.h>

// ---------------------------------------------------------------------------
// Problem constants (from reference)
// ---------------------------------------------------------------------------
#define NNODES 12000
#define NEDGES 192000
#define CH     128
#define WUSED  375           // 12000 bits = 375 dwords
#define WROW   384           // padded row stride (dwords) for 128B alignment

typedef float v2f __attribute__((ext_vector_type(2)));
typedef float v8f __attribute__((ext_vector_type(8)));

__device__ __forceinline__ void atomic_add_f32(float* p, float v) {
  __hip_atomic_fetch_add(p, v, __ATOMIC_RELAXED, __HIP_MEMORY_SCOPE_AGENT);
}
__device__ __forceinline__ void atomic_or_u32(unsigned* p, unsigned v) {
  __hip_atomic_fetch_or(p, v, __ATOMIC_RELAXED, __HIP_MEMORY_SCOPE_AGENT);
}

// f32 WMMA, D = A(16x4) * B(4x16) + C(16x16)
__device__ __forceinline__ v8f wmma_k4(v2f a, v2f b, v8f c) {
  return __builtin_amdgcn_wmma_f32_16x16x4_f32(
      /*neg_a=*/false, a, /*neg_b=*/false, b,
      /*c_mod=*/(short)0, c, /*reuse_a=*/false, /*reuse_b=*/false);
}

// ---------------------------------------------------------------------------
// K0: zero the workspace (agg1, agg2, z1, z2, adjA bits, adjB bits)
// ---------------------------------------------------------------------------
__global__ void k_zero(uint4* ws, long long n4) {
  long long i = (long long)blockIdx.x * blockDim.x + threadIdx.x;
  long long stride = (long long)gridDim.x * blockDim.x;
  uint4 z = make_uint4(0u, 0u, 0u, 0u);
  for (; i < n4; i += stride) ws[i] = z;
}

// ---------------------------------------------------------------------------
// K1: per-edge scatter. One wave per edge:
//   agg1[s][:] += x[t][:]   (float atomics, lane handles 4 channels)
//   adjA[s] bit t set       (lane 0, atomic OR)
// ---------------------------------------------------------------------------
__global__ void k_scatter(const float* __restrict__ x,
                          const int* __restrict__ ei,
                          float* __restrict__ agg1,
                          unsigned* __restrict__ adjA) {
  int gid  = blockIdx.x * blockDim.x + threadIdx.x;
  int e    = gid >> 5;
  int lane = gid & 31;
  if (e >= NEDGES) return;
  int s = ei[e];            // edge_index[0][e]
  int t = ei[NEDGES + e];   // edge_index[1][e]
  if (lane == 0) atomic_or_u32(&adjA[(long long)s * WROW + (t >> 5)], 1u << (t & 31));
  const float4 v = ((const float4*)x)[(long long)t * (CH / 4) + lane];
  float* dst = &agg1[(long long)s * CH + lane * 4];
  atomic_add_f32(dst + 0, v.x);
  atomic_add_f32(dst + 1, v.y);
  atomic_add_f32(dst + 2, v.z);
  atomic_add_f32(dst + 3, v.w);
}

// ---------------------------------------------------------------------------
// K2: 2-hop bitset OR. One wave per edge (s,t): adjB[s] |= adjA[t].
// Skip zero words (rows are ~16/12000 dense) to keep atomic traffic ~E*deg.
// ---------------------------------------------------------------------------
__global__ void k_twohop(const int* __restrict__ ei,
                         const unsigned* __restrict__ adjA,
                         unsigned* __restrict__ adjB) {
  int gid  = blockIdx.x * blockDim.x + threadIdx.x;
  int e    = gid >> 5;
  int lane = gid & 31;
  if (e >= NEDGES) return;
  int s = ei[e];
  int t = ei[NEDGES + e];
  const unsigned* src = &adjA[(long long)t * WROW];
  unsigned* dst = &adjB[(long long)s * WROW];
  for (int w = lane; w < WUSED; w += 32) {
    unsigned word = src[w];
    if (word) atomic_or_u32(&dst[w], word);
  }
}

// ---------------------------------------------------------------------------
// K3: agg2[i] = sum over set bits k of adjB[i] (diag cleared) of x[k].
// One wave per node; lane owns 4 channels (float4); bit scan is wave-uniform.
// ---------------------------------------------------------------------------
__global__ void k_gather2(const float* __restrict__ x,
                          const unsigned* __restrict__ adjB,
                          float* __restrict__ agg2) {
  int gid  = blockIdx.x * blockDim.x + threadIdx.x;
  int i    = gid >> 5;
  int lane = gid & 31;
  if (i >= NNODES) return;
  const unsigned* row = &adjB[(long long)i * WROW];
  const float4* x4 = (const float4*)x;
  const int selfw = i >> 5;
  const unsigned selfbit = 1u << (i & 31);
  float4 acc = make_float4(0.f, 0.f, 0.f, 0.f);
  for (int w = 0; w < WUSED; ++w) {
    unsigned word = row[w];                 // uniform load (broadcast)
    if (w == selfw) word &= ~selfbit;       // zero diagonal
    while (word) {
      int b = __ffs(word) - 1;
      word &= word - 1;
      int k = (w << 5) + b;
      const float4 v = x4[(long long)k * (CH / 4) + lane];
      acc.x += v.x; acc.y += v.y; acc.z += v.z; acc.w += v.w;
    }
  }
  ((float4*)agg2)[(long long)i * (CH / 4) + lane] = acc;
}

// ---------------------------------------------------------------------------
// K4: Z = A(12000x128) @ W(128x128) + bias, via V_WMMA_F32_16X16X4_F32.
// One wave per 16x16 output tile, 32 WMMAs (K=128 in steps of 4).
// ---------------------------------------------------------------------------
__global__ void k_gemm_bias(const float* __restrict__ A,
                            const float* __restrict__ Wm,
                            const float* __restrict__ bias,
                            float* __restrict__ Z) {
  int wave = (blockIdx.x * blockDim.x + threadIdx.x) >> 5;
  int lane = threadIdx.x & 31;
  const int tiles = (NNODES / 16) * (CH / 16);
  if (wave >= tiles) return;
  int tr = wave >> 3;              // row tile (CH/16 == 8)
  int tc = wave & 7;               // col tile
  int half = lane >> 4;            // K-pair selector for A/B fragments
  int m    = lane & 15;            // row (A) / col (B,C,D) within tile
  int row0 = tr * 16;
  int col  = tc * 16 + m;

  v8f c;
  float bv = bias[col];
  #pragma unroll
  for (int r = 0; r < 8; ++r) c[r] = bv;

  const float* arow = &A[(long long)(row0 + m) * CH];
  #pragma unroll 4
  for (int kk = 0; kk < CH; kk += 4) {
    int ka = kk + 2 * half;
    v2f a, b;
    a.x = arow[ka];
    a.y = arow[ka + 1];
    b.x = Wm[(long long)ka * CH + col];
    b.y = Wm[(long long)(ka + 1) * CH + col];
    c = wmma_k4(a, b, c);
  }
  #pragma unroll
  for (int r = 0; r < 8; ++r)
    Z[(long long)(row0 + r + 8 * half) * CH + col] = c[r];
}

// ---------------------------------------------------------------------------
// K5: fused gate GEMM + blend.
//   g = sigmoid([z1|z2](12000x256) @ Wg(256x128) + bg)
//   out = g*z1 + (1-g)*z2
// ---------------------------------------------------------------------------
__global__ void k_gate_blend(const float* __restrict__ z1,
                             const float* __restrict__ z2,
                             const float* __restrict__ Wg,
                             const float* __restrict__ bg,
                             float* __restrict__ out) {
  int wave = (blockIdx.x * blockDim.x + threadIdx.x) >> 5;
  int lane = threadIdx.x & 31;
  const int tiles = (NNODES / 16) * (CH / 16);
  if (wave >= tiles) return;
  int tr = wave >> 3;
  int tc = wave & 7;
  int half = lane >> 4;
  int m    = lane & 15;
  int row0 = tr * 16;
  int col  = tc * 16 + m;

  v8f c;
  float bv = bg[col];
  #pragma unroll
  for (int r = 0; r < 8; ++r) c[r] = bv;

  const float* a1 = &z1[(long long)(row0 + m) * CH];
  const float* a2 = &z2[(long long)(row0 + m) * CH];
  #pragma unroll 4
  for (int kk = 0; kk < 2 * CH; kk += 4) {
    int ka = kk + 2 * half;
    const float* arow = (ka < CH) ? a1 : a2;
    int kloc = (ka < CH) ? ka : (ka - CH);
    v2f a, b;
    a.x = arow[kloc];
    a.y = arow[kloc + 1];
    b.x = Wg[(long long)ka * CH + col];
    b.y = Wg[(long long)(ka + 1) * CH + col];
    c = wmma_k4(a, b, c);
  }
  #pragma unroll
  for (int r = 0; r < 8; ++r) {
    long long idx = (long long)(row0 + r + 8 * half) * CH + col;
    float g = 1.0f / (1.0f + __expf(-c[r]));
    out[idx] = g * z1[idx] + (1.0f - g) * z2[idx];
  }
}

// ---------------------------------------------------------------------------
// Host-side launch
// ---------------------------------------------------------------------------
extern "C" void kernel_launch(void* const* d_in, const int* in_sizes, int n_in,
                              void* d_out, int out_size, void* d_ws, size_t ws_size,
                              hipStream_t stream) {
  const float* x   = (const float*)d_in[0];
  const int*   ei  = (const int*)d_in[1];   // (2,E) int32
  const float* W1  = (const float*)d_in[2];
  const float* b1  = (const float*)d_in[3];
  const float* W2  = (const float*)d_in[4];
  const float* b2  = (const float*)d_in[5];
  const float* Wg  = (const float*)d_in[6];
  const float* bg  = (const float*)d_in[7];
  float* out = (float*)d_out;

  // workspace layout (all 16B-aligned)
  float*    agg1 = (float*)d_ws;
  float*    agg2 = agg1 + (long long)NNODES * CH;
  float*    z1   = agg2 + (long long)NNODES * CH;
  float*    z2   = z1   + (long long)NNODES * CH;
  unsigned* adjA = (unsigned*)(z2 + (long long)NNODES * CH);
  unsigned* adjB = adjA + (long long)NNODES * WROW;

  const long long total_dwords = 4LL * NNODES * CH + 2LL * NNODES * WROW;
  const long long n4 = total_dwords / 4;

  k_zero<<<2048, 256, 0, stream>>>((uint4*)d_ws, n4);

  const int edge_threads = NEDGES * 32;
  k_scatter<<<(edge_threads + 255) / 256, 256, 0, stream>>>(x, ei, agg1, adjA);
  k_twohop<<<(edge_threads + 255) / 256, 256, 0, stream>>>(ei, adjA, adjB);

  const int node_threads = NNODES * 32;
  k_gather2<<<(node_threads + 255) / 256, 256, 0, stream>>>(x, adjB, agg2);

  const int tiles = (NNODES / 16) * (CH / 16);       // 6000 waves
  const int gemm_blocks = (tiles * 32 + 255) / 256;  // 750 blocks
  k_gemm_bias<<<gemm_blocks, 256, 0, stream>>>(agg1, W1, b1, z1);
  k_gemm_bias<<<gemm_blocks, 256, 0, stream>>>(agg2, W2, b2, z2);
  k_gate_blend<<<gemm_blocks, 256, 0, stream>>>(z1, z2, Wg, bg, out);
}